// GIN_tree_60971355734188
// MI455X (gfx1250) — compile-verified
//
#include <hip/hip_runtime.h>

#define NN 300000   // nodes
#define NE 600000   // edges
#define NG 10000    // graphs
#define H  128      // hidden

typedef __attribute__((ext_vector_type(2))) float v2f;
typedef __attribute__((ext_vector_type(4))) float v4f;
typedef __attribute__((ext_vector_type(8))) float v8f;

// ---------------- init agg + out to -inf ----------------
__global__ void k_init(float* __restrict__ agg, float* __restrict__ out) {
  long i = (long)blockIdx.x * blockDim.x + threadIdx.x;
  const float ninf = -__builtin_inff();
  if (i < (long)NN * H) agg[i] = ninf;
  if (i < (long)NG * H) out[i] = ninf;
}

// ---------------- Wt[k][j] = W[j][k] ----------------
__global__ void k_transpose(const float* __restrict__ W, float* __restrict__ Wt) {
  int i = blockIdx.x * blockDim.x + threadIdx.x;   // 16384 elements
  int j = i >> 7;
  int k = i & 127;
  Wt[k * H + j] = W[i];
}

// ---------------- edge scatter-max: agg[dst] = max(emb[wid[src]]) ----------------
__global__ void k_edge(const int* __restrict__ wid, const int* __restrict__ src,
                       const int* __restrict__ dst, const float* __restrict__ emb,
                       float* __restrict__ agg) {
  long t = (long)blockIdx.x * blockDim.x + threadIdx.x;
  int e = (int)(t >> 5);
  if (e >= NE) return;
  int c = ((int)t & 31) * 4;                       // 32 lanes cover 128 dims
  int s = src[e];
  int d = dst[e];
  const v4f v = *(const v4f*)(emb + (long)wid[s] * H + c);
  float* a = agg + (long)d * H + c;
  atomicMax(a + 0, v.x);                           // global_atomic_max_num_f32
  atomicMax(a + 1, v.y);
  atomicMax(a + 2, v.z);
  atomicMax(a + 3, v.w);
}

// ---------------- rst = h + where(finite(agg), agg, 0), in place ----------------
__global__ void k_rst(const int* __restrict__ wid, const float* __restrict__ emb,
                      float* __restrict__ agg) {
  long t = (long)blockIdx.x * blockDim.x + threadIdx.x;
  int n = (int)(t >> 5);
  if (n >= NN) return;
  int c = ((int)t & 31) * 4;
  v4f hv = *(const v4f*)(emb + (long)wid[n] * H + c);
  v4f* ap = (v4f*)(agg + (long)n * H + c);
  v4f a = *ap;
  const float ninf = -__builtin_inff();
  a.x = hv.x + (a.x == ninf ? 0.0f : a.x);
  a.y = hv.y + (a.y == ninf ? 0.0f : a.y);
  a.z = hv.z + (a.z == ninf ? 0.0f : a.z);
  a.w = hv.w + (a.w == ninf ? 0.0f : a.w);
  *ap = a;
}

// ---------------- fused GEMM (WMMA f32 16x16x4) + bias + per-graph max readout ----
// 8 waves / block, each wave computes a 16-node x 128-col tile of res = rst @ W^T + b
// and atomically max-reduces it into out[graph_ids[node]].
#define LDS_PAD 132   // floats per LDS row: 132*4 bytes -> 16B aligned rows, bank-safe

__global__ __launch_bounds__(256) void k_gemm(const float* __restrict__ rst,
                                              const float* __restrict__ Wt,
                                              const float* __restrict__ bias,
                                              const int* __restrict__ gids,
                                              float* __restrict__ out) {
  __shared__ float lds[8 * 16 * LDS_PAD];
  const int wave = threadIdx.x >> 5;
  const int lane = threadIdx.x & 31;
  const int n0 = (blockIdx.x * 8 + wave) * 16;     // 16-node tile base
  if (n0 >= NN) return;                            // whole-wave guard (EXEC stays all-1)

  float* tile = lds + wave * 16 * LDS_PAD;

  // Stage 16x128 rst tile into LDS, coalesced (32 lanes x float4 per row).
#pragma unroll
  for (int r = 0; r < 16; ++r) {
    v4f v = *(const v4f*)(rst + (long)(n0 + r) * H + lane * 4);
    *(v4f*)(tile + r * LDS_PAD + lane * 4) = v;
  }

  // WMMA fragment coordinates (f32 16x16x4 layouts, ISA 7.12.2):
  // A: lanes 0-15 -> M=lane, K={k0,k0+1}; lanes 16-31 -> M=lane-16, K={k0+2,k0+3}
  // B: VGPR0 -> K=k0(+2 for hi lanes), N=lane&15; VGPR1 -> K+1
  const int rowA = lane & 15;
  const int koff = (lane >> 4) * 2;

  v8f acc[8] = {};
#pragma unroll 4
  for (int k0 = 0; k0 < H; k0 += 4) {
    v2f a = *(const v2f*)(tile + rowA * LDS_PAD + k0 + koff);
#pragma unroll
    for (int t = 0; t < 8; ++t) {
      v2f bf;
      bf.x = Wt[(k0 + koff) * H + t * 16 + rowA];
      bf.y = Wt[(k0 + koff + 1) * H + t * 16 + rowA];
      acc[t] = __builtin_amdgcn_wmma_f32_16x16x4_f32(
          /*neg_a=*/false, a, /*neg_b=*/false, bf,
          /*c_mod=*/(short)0, acc[t], /*reuse_a=*/false, /*reuse_b=*/false);
    }
  }

  // C/D layout: VGPR v, lanes 0-15 -> M=v, N=lane; lanes 16-31 -> M=v+8, N=lane-16
  int g8[8];
#pragma unroll
  for (int v = 0; v < 8; ++v) g8[v] = gids[n0 + v + (lane >> 4) * 8];

#pragma unroll
  for (int t = 0; t < 8; ++t) {
    const int col = t * 16 + rowA;
    const float bv = bias[col];
#pragma unroll
    for (int v = 0; v < 8; ++v) {
      atomicMax(out + (long)g8[v] * H + col, acc[t][v] + bv);
    }
  }
}

extern "C" void kernel_launch(void* const* d_in, const int* in_sizes, int n_in,
                              void* d_out, int out_size, void* d_ws, size_t ws_size,
                              hipStream_t stream) {
  const int*   wid  = (const int*)d_in[0];
  const int*   src  = (const int*)d_in[1];
  const int*   dst  = (const int*)d_in[2];
  const int*   gids = (const int*)d_in[3];
  const float* emb  = (const float*)d_in[4];
  const float* W    = (const float*)d_in[5];
  const float* b    = (const float*)d_in[6];
  float* out = (float*)d_out;

  // workspace layout: agg/rst [NN*H] f32, then Wt [H*H] f32  (~153.7 MB)
  float* agg = (float*)d_ws;
  float* Wt  = agg + (size_t)NN * H;

  {
    long n = (long)NN * H;                               // 38.4M (covers out too)
    int blocks = (int)((n + 255) / 256);
    k_init<<<blocks, 256, 0, stream>>>(agg, out);
  }
  k_transpose<<<(H * H + 255) / 256, 256, 0, stream>>>(W, Wt);
  {
    long n = (long)NE * 32;
    int blocks = (int)((n + 255) / 256);
    k_edge<<<blocks, 256, 0, stream>>>(wid, src, dst, emb, agg);
  }
  {
    long n = (long)NN * 32;
    int blocks = (int)((n + 255) / 256);
    k_rst<<<blocks, 256, 0, stream>>>(wid, emb, agg);
  }
  {
    int blocks = (NN + 127) / 128;                       // 8 waves x 16 nodes each
    k_gemm<<<blocks, 256, 0, stream>>>(agg, Wt, b, gids, out);
  }
}